// LocallyNonlinear_12515534700891
// MI455X (gfx1250) — compile-verified
//
#include <hip/hip_runtime.h>
#include <math.h>

// ---------------------------------------------------------------------------
// LocallyNonlinear on gfx1250 (MI455X).
// Heavy path: 3->56-plane 7x7 conv as implicit GEMM via V_WMMA_F32_16X16X4_F32
// (K=147 padded to 148 = 37 steps of 4; N=56 padded to 64 = 4 tiles of 16).
// A-fragment gather offsets come from a precomputed packed-u16 LDS table, so
// the inner loop is pure ds_load + wmma with a handful of VALU ops.
// ---------------------------------------------------------------------------

typedef __attribute__((ext_vector_type(2))) float v2f;
typedef __attribute__((ext_vector_type(8))) float v8f;

#define BATCH   8
#define ICH     3
#define HH      256
#define WW      256
#define KTOT    147      // 3 * 49
#define KPAD    148
#define KSTEPS  37       // 148 / 4
#define NPLANES 56       // 8 (o0 real) + 48 (t re/im)
#define NTILES  4        // 64 columns
#define TILE_W  128      // pixels per block = 8 waves * 16
#define TSTRIDE 136      // input halo tile row stride (cols 0..133 data, 134/135 zero)
#define CSTRIDE 68       // C-dump pixel stride (bank spread)
#define NBLOCKS (BATCH * HH * 2)   // 4096 conv blocks

// workspace layout (float offsets)
#define WS_STAT 0                          // 48: sum[24], sumsq[24]
#define WS_W1   48                         // 48: W1eff complex [g][i]
#define WS_BPK  96                         // 37*4*32*2 = 9472 floats (B frags)
#define WS_ATAB (WS_BPK + KSTEPS*NTILES*32*2)   // 74 dwords: packed u16 A offsets
#define WS_BLK  (WS_ATAB + 74)             // 4096*48 block partials
#define WS_VBUF (WS_BLK + NBLOCKS*48)      // B*24*H*W floats

#define PI_F 3.14159265358979323846f

// ---------------------------------------------------------------------------
// Kernel 0: collapse tiled weights; emit B matrix pre-swizzled in WMMA
// fragment order, plus packed A-gather offset table.
// frag fi=(k0*4+t)*32+lane holds {B[4k0+2h][n], B[4k0+2h+1][n]},
// h=lane>>4, n=t*16+(lane&15).
// ---------------------------------------------------------------------------
__global__ __launch_bounds__(256)
void ln_prep_kernel(const float* __restrict__ w0,   // (8,12,7,7) c64
                    const float* __restrict__ w1,   // (8,12,1,1) c64
                    const float* __restrict__ w2a,  // (24,12,7,7) c64
                    float* __restrict__ ws)
{
    const int tid = threadIdx.x;
    if (tid < 24) {                         // W1eff[g][i] = sum_m w1[g,3m+i]
        int g = tid / 3, i = tid % 3;
        float re = 0.f, im = 0.f;
        for (int m = 0; m < 4; ++m) {
            int ci = g * 12 + 3 * m + i;
            re += w1[2 * ci];
            im += w1[2 * ci + 1];
        }
        ws[WS_W1 + 2 * tid]     = re;
        ws[WS_W1 + 2 * tid + 1] = im;
    }
    // packed u16 A-offset table: entry i packs off(2i) | off(2i+1)<<16
    if (tid < KPAD / 2) {
        unsigned pk = 0;
        for (int j = 0; j < 2; ++j) {
            int k = 2 * tid + j;
            unsigned off;
            if (k < KTOT) {
                int c = k / 49, r = k % 49;
                int dy = r / 7, dx = r % 7;
                off = (unsigned)(((c * 7 + dy) * TSTRIDE + dx) * 4);
            } else {
                off = (unsigned)(135 * 4);   // k >= 147 pad; its B rows are zero
            }
            pk |= off << (16 * j);
        }
        ((unsigned*)ws)[WS_ATAB + tid] = pk;
    }
    // B fragments
    for (int idx = tid; idx < KSTEPS * NTILES * 32; idx += 256) {
        int lane = idx & 31;
        int t    = (idx >> 5) & 3;
        int k0   = idx >> 7;
        int n    = t * 16 + (lane & 15);
        int kb   = k0 * 4 + 2 * (lane >> 4);
        float out0 = 0.f, out1 = 0.f;
        for (int j = 0; j < 2; ++j) {
            int k = kb + j;
            float s = 0.f;
            if (k < KTOT && n < NPLANES) {
                int c = k / 49, r = k % 49;
                if (n < 8) {                // o0 real planes
                    for (int m = 0; m < 4; ++m)
                        s += w0[2 * ((n * 12 + 3 * m + c) * 49 + r)];
                } else {                    // t re/im planes
                    int ch = (n - 8) >> 1, im = (n - 8) & 1;
                    for (int m = 0; m < 4; ++m)
                        s += w2a[2 * ((ch * 12 + 3 * m + c) * 49 + r) + im];
                }
            }
            if (j == 0) out0 = s; else out1 = s;
        }
        ws[WS_BPK + 2 * idx]     = out0;
        ws[WS_BPK + 2 * idx + 1] = out1;
    }
}

// ---------------------------------------------------------------------------
// Kernel 1: WMMA conv + fused nonlinear epilogue + deterministic BN partials.
// Grid (2, 256, 8): x-tile, row, batch. 256 threads = 8 waves, 16 px each.
// ---------------------------------------------------------------------------
__global__ __launch_bounds__(256)
void ln_conv_kernel(const float* __restrict__ x,
                    const float* __restrict__ b0,    // (8)  c64
                    const float* __restrict__ b1,    // (8)  c64
                    const float* __restrict__ b2a,   // (24) c64
                    const float* __restrict__ w2b,   // (8,3) c64
                    const float* __restrict__ b2b,   // (8)  c64
                    float* __restrict__ ws)
{
    __shared__ __align__(16) float tileIn[ICH * 7 * TSTRIDE];          // 11424 B
    __shared__ __align__(16) float shB[KSTEPS * NTILES * 32 * 2];      // 37888 B (reused as C dump)
    __shared__ __align__(16) float vtmp[24 * TILE_W];                  // 12288 B
    __shared__ __align__(8)  unsigned shATab[KPAD / 2];                //   296 B

    const int tid = threadIdx.x;
    const int bx  = blockIdx.x;            // 0..1
    const int by  = blockIdx.y;            // row
    const int bz  = blockIdx.z;            // batch
    const int x0  = bx * TILE_W;

    // ---- stage input halo tile (zero 'SAME' padding; cols>=134 zero) ----
    for (int idx = tid; idx < ICH * 7 * TSTRIDE; idx += 256) {
        int col = idx % TSTRIDE;
        int tmp = idx / TSTRIDE;
        int row = tmp % 7;
        int c   = tmp / 7;
        float v = 0.f;
        int gx = x0 - 3 + col;
        int gy = by - 3 + row;
        if (col < TILE_W + 6 && gx >= 0 && gx < WW && gy >= 0 && gy < HH)
            v = x[((bz * ICH + c) * HH + gy) * WW + gx];
        tileIn[idx] = v;
    }
    // ---- stage pre-swizzled B fragments + A offset table ----
    for (int idx = tid; idx < KSTEPS * NTILES * 32 * 2; idx += 256)
        shB[idx] = ws[WS_BPK + idx];
    if (tid < KPAD / 2)
        shATab[tid] = ((const unsigned*)ws)[WS_ATAB + tid];
    __syncthreads();

    const int lane = tid & 31;
    const int wave = tid >> 5;
    const int half = lane >> 4;
    const int ml   = lane & 15;
    const int px   = wave * 16 + ml;       // pixel for A fragment

    v8f acc0 = {}, acc1 = {}, acc2 = {}, acc3 = {};
    const v2f* bfr = (const v2f*)shB;
    const char* tbase = (const char*)tileIn + px * 4;

    for (int k0 = 0; k0 < KSTEPS; ++k0) {
        // A fragment: lane half h supplies K = 4k0+2h, 4k0+2h+1 for pixel ml.
        unsigned pk = shATab[k0 * 2 + half];
        v2f a;
        a.x = *(const float*)(tbase + (pk & 0xffffu));
        a.y = *(const float*)(tbase + (pk >> 16));

        v2f bf0 = bfr[(k0 * 4 + 0) * 32 + lane];
        v2f bf1 = bfr[(k0 * 4 + 1) * 32 + lane];
        v2f bf2 = bfr[(k0 * 4 + 2) * 32 + lane];
        v2f bf3 = bfr[(k0 * 4 + 3) * 32 + lane];

        acc0 = __builtin_amdgcn_wmma_f32_16x16x4_f32(false, a, false, bf0, (short)0, acc0, false, false);
        acc1 = __builtin_amdgcn_wmma_f32_16x16x4_f32(false, a, false, bf1, (short)0, acc1, false, false);
        acc2 = __builtin_amdgcn_wmma_f32_16x16x4_f32(false, a, false, bf2, (short)0, acc2, false, false);
        acc3 = __builtin_amdgcn_wmma_f32_16x16x4_f32(false, a, false, bf3, (short)0, acc3, false, false);
    }
    __syncthreads();                       // everyone done reading shB

    // ---- dump C tiles: Cd[pixel][plane]; VGPR j -> M=j(+8 for high half) ----
    float* Cd = shB;
    {
        int pbase = wave * 16 + half * 8;
#pragma unroll
        for (int j = 0; j < 8; ++j) {
            Cd[(pbase + j) * CSTRIDE +  0 + ml] = acc0[j];
            Cd[(pbase + j) * CSTRIDE + 16 + ml] = acc1[j];
            Cd[(pbase + j) * CSTRIDE + 32 + ml] = acc2[j];
            Cd[(pbase + j) * CSTRIDE + 48 + ml] = acc3[j];
        }
    }
    __syncthreads();

    // ---- fused nonlinear epilogue: 2 threads per pixel, 4 groups each ----
    {
        const int p     = tid >> 1;        // 0..127
        const int ghalf = tid & 1;
        const int gx    = x0 + p;
        float lr[3], li[3];
#pragma unroll
        for (int c = 0; c < 3; ++c) {
            float xv = tileIn[(c * 7 + 3) * TSTRIDE + p + 3];
            lr[c] = logf(fabsf(xv));
            li[c] = (xv < 0.f) ? PI_F : 0.f;
        }
        const float* W1 = ws + WS_W1;
        float* vbuf = ws + WS_VBUF;
#pragma unroll
        for (int gg = 0; gg < 4; ++gg) {
            int g = ghalf * 4 + gg;
            // o0 (real part only)
            float v0 = Cd[p * CSTRIDE + g] + b0[2 * g];
            // o1 = exp(W1eff . log(x) + b1), real part
            float zr = b1[2 * g], zi = b1[2 * g + 1];
#pragma unroll
            for (int c = 0; c < 3; ++c) {
                float wr = W1[2 * (g * 3 + c)], wi = W1[2 * (g * 3 + c) + 1];
                zr += wr * lr[c] - wi * li[c];
                zi += wr * li[c] + wi * lr[c];
            }
            float v1 = expf(zr) * cosf(zi);
            // o2 = exp(w2b . log(t) + b2b), real part
            float ur = b2b[2 * g], ui = b2b[2 * g + 1];
#pragma unroll
            for (int k = 0; k < 3; ++k) {
                int ch = 3 * g + k;
                float tre = Cd[p * CSTRIDE + 8 + 2 * ch] + b2a[2 * ch];
                float tim = Cd[p * CSTRIDE + 9 + 2 * ch] + b2a[2 * ch + 1];
                float mr = 0.5f * logf(tre * tre + tim * tim);
                float ai = atan2f(tim, tre);
                float wr = w2b[2 * (g * 3 + k)], wi = w2b[2 * (g * 3 + k) + 1];
                ur += wr * mr - wi * ai;
                ui += wr * ai + wi * mr;
            }
            float v2 = expf(ur) * cosf(ui);

            int ch0 = 3 * g;
            vbuf[((bz * 24 + ch0 + 0) * HH + by) * WW + gx] = v0;
            vbuf[((bz * 24 + ch0 + 1) * HH + by) * WW + gx] = v1;
            vbuf[((bz * 24 + ch0 + 2) * HH + by) * WW + gx] = v2;
            vtmp[(ch0 + 0) * TILE_W + p] = v0;
            vtmp[(ch0 + 1) * TILE_W + p] = v1;
            vtmp[(ch0 + 2) * TILE_W + p] = v2;
        }
    }
    __syncthreads();

    // ---- deterministic per-block BN partial sums ----
    if (tid < 48) {
        int ch   = tid % 24;
        int issq = tid / 24;
        float s = 0.f;
        for (int p = 0; p < TILE_W; ++p) {
            float v = vtmp[ch * TILE_W + p];
            s += issq ? v * v : v;
        }
        int blk = (bz * 256 + by) * 2 + bx;
        ws[WS_BLK + blk * 48 + tid] = s;
    }
}

// ---------------------------------------------------------------------------
// Kernel 2: fixed-order reduction of per-block partials -> BN stats.
// ---------------------------------------------------------------------------
__global__ void ln_stats_kernel(float* __restrict__ ws)
{
    int tid = threadIdx.x;
    if (tid >= 48) return;
    float s = 0.f;
    for (int blk = 0; blk < NBLOCKS; ++blk)
        s += ws[WS_BLK + blk * 48 + tid];
    ws[WS_STAT + tid] = s;
}

// ---------------------------------------------------------------------------
// Kernel 3: BatchNorm (training stats) + grouped 1x1 agg + sigmoid.
// ---------------------------------------------------------------------------
__global__ __launch_bounds__(256)
void ln_out_kernel(const float* __restrict__ ws,
                   const float* __restrict__ gamma,
                   const float* __restrict__ beta,
                   const float* __restrict__ aggw,  // (8,3)
                   const float* __restrict__ aggb,  // (8)
                   float* __restrict__ out)
{
    int idx = blockIdx.x * 256 + threadIdx.x;
    if (idx >= BATCH * 8 * HH * WW) return;
    int hw = idx & 65535;
    int g  = (idx >> 16) & 7;
    int b  = idx >> 19;
    const float* vbuf = ws + WS_VBUF;
    const float invN = 1.f / (float)(BATCH * HH * WW);
    float acc = aggb[g];
#pragma unroll
    for (int k = 0; k < 3; ++k) {
        int ch = 3 * g + k;
        float mean = ws[WS_STAT + ch] * invN;
        float var  = ws[WS_STAT + 24 + ch] * invN - mean * mean;
        float vv   = vbuf[(b * 24 + ch) * 65536 + hw];
        float xn   = (vv - mean) * rsqrtf(var + 1e-5f);
        acc += aggw[g * 3 + k] * (xn * gamma[ch] + beta[ch]);
    }
    out[idx] = 1.f / (1.f + expf(-acc));
}

// ---------------------------------------------------------------------------
extern "C" void kernel_launch(void* const* d_in, const int* in_sizes, int n_in,
                              void* d_out, int out_size, void* d_ws, size_t ws_size,
                              hipStream_t stream)
{
    const float* x    = (const float*)d_in[0];
    const float* w0   = (const float*)d_in[1];
    const float* b0   = (const float*)d_in[2];
    const float* w1   = (const float*)d_in[3];
    const float* b1   = (const float*)d_in[4];
    const float* w2a  = (const float*)d_in[5];
    const float* b2a  = (const float*)d_in[6];
    const float* w2b  = (const float*)d_in[7];
    const float* b2b  = (const float*)d_in[8];
    const float* bng  = (const float*)d_in[9];
    const float* bnb  = (const float*)d_in[10];
    const float* aggw = (const float*)d_in[11];
    const float* aggb = (const float*)d_in[12];
    float* ws  = (float*)d_ws;
    float* out = (float*)d_out;

    ln_prep_kernel<<<1, 256, 0, stream>>>(w0, w1, w2a, ws);

    dim3 grid(2, HH, BATCH);
    ln_conv_kernel<<<grid, 256, 0, stream>>>(x, b0, b1, b2a, w2b, b2b, ws);

    ln_stats_kernel<<<1, 64, 0, stream>>>(ws);

    int nout = BATCH * 8 * HH * WW;
    ln_out_kernel<<<(nout + 255) / 256, 256, 0, stream>>>(ws, bng, bnb, aggw, aggb, out);
}